// SimilarityTreeLSTM_29154238005600
// MI455X (gfx1250) — compile-verified
//
#include <hip/hip_runtime.h>
#include <hip/hip_bf16.h>
#include <math.h>

typedef __attribute__((ext_vector_type(2))) float v2f;
typedef __attribute__((ext_vector_type(8))) float v8f;

#define NLEAF 2048
#define NNODE 4095      // 2*2048-1
#define MEMD  150
#define INDIM 300
#define IOU   450

#if defined(__has_builtin)
#if __has_builtin(__builtin_amdgcn_wmma_f32_16x16x4_f32)
#define HAVE_WMMA_F32 1
#endif
#endif
#ifndef HAVE_WMMA_F32
#define HAVE_WMMA_F32 0
#endif

__device__ __forceinline__ float sigf(float x) { return 1.0f / (1.0f + expf(-x)); }

// ---------------- B-matrix packing (zero-padded for unguarded WMMA loads) ---

// W_ioux (300x450 row-major) -> B (300 x 464), cols >=450 zero
__global__ void pack_b_leaf(const float* __restrict__ W, float* __restrict__ B) {
  int idx = blockIdx.x * blockDim.x + threadIdx.x;
  if (idx >= 300 * 464) return;
  int k = idx / 464, n = idx % 464;
  B[idx] = (n < 450) ? W[k * 450 + n] : 0.0f;
}

// [W_iouh (150x450) | W_fh (150x150)] -> B (152 x 608), padded with zeros
__global__ void pack_b_big(const float* __restrict__ Wiouh, const float* __restrict__ Wfh,
                           float* __restrict__ B) {
  int idx = blockIdx.x * blockDim.x + threadIdx.x;
  if (idx >= 152 * 608) return;
  int k = idx / 608, n = idx % 608;
  float v = 0.0f;
  if (k < 150) {
    if (n < 450)       v = Wiouh[k * 450 + n];
    else if (n < 600)  v = Wfh[k * 150 + (n - 450)];
  }
  B[idx] = v;
}

// ---------------- Leaf stage ------------------------------------------------

// A[row, 0:300] = emb[tokens]; rows = [tree0 2048 | tree1 2048]
__global__ void leaf_gather(const int* __restrict__ lt, const int* __restrict__ rt,
                            const float* __restrict__ emb, float* __restrict__ A) {
  int idx = blockIdx.x * blockDim.x + threadIdx.x;
  if (idx >= 4096 * INDIM) return;
  int row = idx / INDIM, c = idx % INDIM;
  int t = row >> 11, m = row & 2047;
  int tok = t ? rt[m] : lt[m];
  A[idx] = emb[tok * INDIM + c];
}

// ---------------- WMMA f32 GEMM: D = A(MxK) * B(KxN) ------------------------
// one wave per 16x16 tile; A zero-padded so K loop is unguarded
__global__ __launch_bounds__(32) void wmma_gemm_f32(
    const float* __restrict__ A, int lda,
    const float* __restrict__ B, int ldb,
    float* __restrict__ D, int ldc,
    int K, int Mvalid, int Nvalid) {
  int tm = blockIdx.x, tn = blockIdx.y;
  int lane = threadIdx.x;
  int half = lane >> 4;       // 0: K pair {k,k+1}; 1: {k+2,k+3}
  int l16  = lane & 15;
#if HAVE_WMMA_F32
  v8f acc = {};
  const float* arow = A + (tm * 16 + l16) * (long)lda;
  const float* bcol = B + tn * 16 + l16;
  for (int k = 0; k < K; k += 4) {
    int ka = k + half * 2;
    v2f a, b;
    a.x = arow[ka];
    a.y = arow[ka + 1];
    b.x = bcol[(long)ka * ldb];
    b.y = bcol[(long)(ka + 1) * ldb];
    acc = __builtin_amdgcn_wmma_f32_16x16x4_f32(false, a, false, b,
                                                (short)0, acc, false, false);
  }
  int col = tn * 16 + l16;
  int rbase = tm * 16 + half * 8;
  if (col < Nvalid) {
#pragma unroll
    for (int r = 0; r < 8; ++r) {
      int row = rbase + r;
      if (row < Mvalid) D[(long)row * ldc + col] = acc[r];
    }
  }
#else
  // scalar fallback with identical output mapping
  int col = tn * 16 + l16;
  int rbase = tm * 16 + half * 8;
  if (col < Nvalid) {
    for (int r = 0; r < 8; ++r) {
      int row = rbase + r;
      if (row >= Mvalid) continue;
      float s = 0.0f;
      for (int k = 0; k < K; ++k) s += A[(long)row * lda + k] * B[(long)k * ldb + col];
      D[(long)row * ldc + col] = s;
    }
  }
#endif
}

// ---------------- Combine kernels ------------------------------------------

__global__ void leaf_combine(const float* __restrict__ P,
                             const float* __restrict__ b_ioux, const float* __restrict__ b_iouh,
                             float* __restrict__ C0, float* __restrict__ H0,
                             float* __restrict__ C1, float* __restrict__ H1) {
  int idx = blockIdx.x * blockDim.x + threadIdx.x;
  if (idx >= 2 * NLEAF * MEMD) return;
  int j = idx % MEMD;
  int rm = idx / MEMD;           // row in P = t*2048+m
  int m = rm & 2047, t = rm >> 11;
  const float* p = P + (long)rm * 464;
  float pi = p[j]         + b_ioux[j]         + b_iouh[j];
  float po = p[150 + j]   + b_ioux[150 + j]   + b_iouh[150 + j];
  float pu = p[300 + j]   + b_ioux[300 + j]   + b_iouh[300 + j];
  float i = sigf(pi), o = sigf(po), u = tanhf(pu);
  float c = i * u;
  float h = o * tanhf(c);
  float* C = t ? C1 : C0;
  float* H = t ? H1 : H0;
  C[m * MEMD + j] = c;
  H[m * MEMD + j] = h;
}

// rows of A: [t0-left M | t0-right M | t1-left M | t1-right M], zero pad to rowsPad x 152
__global__ void level_gather(const float* __restrict__ H0, const float* __restrict__ H1,
                             const int* __restrict__ li, const int* __restrict__ ri,
                             float* __restrict__ A, int off, int M, int rowsPad) {
  int idx = blockIdx.x * blockDim.x + threadIdx.x;
  if (idx >= rowsPad * 152) return;
  int row = idx / 152, c = idx % 152;
  float v = 0.0f;
  if (row < 4 * M && c < MEMD) {
    int seg = row / M, m = row % M;
    int t = seg >> 1, side = seg & 1;
    int child = side ? ri[off + m] : li[off + m];
    const float* H = t ? H1 : H0;
    v = H[child * MEMD + c];
  }
  A[idx] = v;
}

__global__ void level_combine(const float* __restrict__ P,
                              const int* __restrict__ li, const int* __restrict__ ri,
                              const float* __restrict__ b_iouh, const float* __restrict__ b_fh,
                              float* __restrict__ C0, float* __restrict__ H0,
                              float* __restrict__ C1, float* __restrict__ H1,
                              int off, int M) {
  int idx = blockIdx.x * blockDim.x + threadIdx.x;
  if (idx >= 2 * M * MEMD) return;
  int j = idx % MEMD;
  int tm = idx / MEMD;
  int m = tm % M, t = tm / M;
  const float* pl = P + (long)((2 * t) * M + m) * 608;
  const float* pr = P + (long)((2 * t + 1) * M + m) * 608;
  float pi  = pl[j]       + pr[j]       + b_iouh[j];
  float po  = pl[150 + j] + pr[150 + j] + b_iouh[150 + j];
  float pu  = pl[300 + j] + pr[300 + j] + b_iouh[300 + j];
  float pfl = pl[450 + j] + b_fh[j];
  float pfr = pr[450 + j] + b_fh[j];
  float i = sigf(pi), o = sigf(po), u = tanhf(pu);
  float fl = sigf(pfl), fr = sigf(pfr);
  float* C = t ? C1 : C0;
  float* H = t ? H1 : H0;
  int lc = li[off + m], rc = ri[off + m];
  float c = i * u + fl * C[lc * MEMD + j] + fr * C[rc * MEMD + j];
  float h = o * tanhf(c);
  int k = NLEAF + off + m;
  C[k * MEMD + j] = c;
  H[k * MEMD + j] = h;
}

// ---------------- Attention (only last row/col of scores needed) -----------

__global__ void attn_scores(const float* __restrict__ H0, const float* __restrict__ H1,
                            float* __restrict__ S) {
  int idx = blockIdx.x * blockDim.x + threadIdx.x;
  if (idx >= 2 * NNODE) return;
  const float *a, *b;
  if (idx < NNODE) { a = H0 + (long)(NNODE - 1) * MEMD; b = H1 + (long)idx * MEMD; }
  else             { a = H0 + (long)(idx - NNODE) * MEMD; b = H1 + (long)(NNODE - 1) * MEMD; }
  float s = 0.0f;
  for (int c = 0; c < MEMD; ++c) s += a[c] * b[c];
  S[idx] = s;
}

// block 0: beta = softmax(S[0:N]) @ H1 ; block 1: alpha = softmax(S[N:2N]) @ H0
__global__ void attn_reduce(const float* __restrict__ S,
                            const float* __restrict__ H0, const float* __restrict__ H1,
                            float* __restrict__ vec) {
  int which = blockIdx.x;
  const float* s = S + which * NNODE;
  const float* H = which ? H0 : H1;
  __shared__ float red[256];
  __shared__ float smax, ssum;
  int tid = threadIdx.x;
  float m = -3.4e38f;
  for (int j = tid; j < NNODE; j += 256) m = fmaxf(m, s[j]);
  red[tid] = m; __syncthreads();
  for (int st = 128; st > 0; st >>= 1) {
    if (tid < st) red[tid] = fmaxf(red[tid], red[tid + st]);
    __syncthreads();
  }
  if (tid == 0) smax = red[0];
  __syncthreads();
  float sum = 0.0f;
  for (int j = tid; j < NNODE; j += 256) sum += expf(s[j] - smax);
  red[tid] = sum; __syncthreads();
  for (int st = 128; st > 0; st >>= 1) {
    if (tid < st) red[tid] += red[tid + st];
    __syncthreads();
  }
  if (tid == 0) ssum = red[0];
  __syncthreads();
  if (tid < MEMD) {
    float acc = 0.0f;
    for (int j = 0; j < NNODE; ++j) acc += expf(s[j] - smax) * H[(long)j * MEMD + tid];
    vec[which * MEMD + tid] = acc / ssum;
  }
}

__global__ void final_mlp(const float* __restrict__ H0, const float* __restrict__ H1,
                          const float* __restrict__ vec,
                          const float* __restrict__ W_attn, const float* __restrict__ b_attn,
                          const float* __restrict__ W_wh, const float* __restrict__ b_wh,
                          const float* __restrict__ W_wp, const float* __restrict__ b_wp,
                          float* __restrict__ out) {
  __shared__ float lh[MEMD], rh[MEMD], feats[2 * MEMD], hid[50], logit[5];
  int tid = threadIdx.x;
  if (tid < MEMD) {
    const float* hl = H0 + (long)(NNODE - 1) * MEMD;
    const float* hr = H1 + (long)(NNODE - 1) * MEMD;
    const float* beta  = vec;
    const float* alpha = vec + MEMD;
    float aL = b_attn[tid], aR = b_attn[tid];
    for (int c = 0; c < MEMD; ++c) {
      float w = W_attn[c * MEMD + tid];
      aL += hl[c] * w;
      aR += hr[c] * w;
    }
    for (int c = 0; c < MEMD; ++c) {
      float w = W_attn[(MEMD + c) * MEMD + tid];
      aL += beta[c] * w;
      aR += alpha[c] * w;
    }
    lh[tid] = aL; rh[tid] = aR;
  }
  __syncthreads();
  if (tid < MEMD) {
    feats[tid]        = lh[tid] * rh[tid];
    feats[MEMD + tid] = fabsf(lh[tid] - rh[tid]);
  }
  __syncthreads();
  if (tid < 50) {
    float a = b_wh[tid];
    for (int c = 0; c < 2 * MEMD; ++c) a += feats[c] * W_wh[c * 50 + tid];
    hid[tid] = sigf(a);
  }
  __syncthreads();
  if (tid < 5) {
    float a = b_wp[tid];
    for (int k = 0; k < 50; ++k) a += hid[k] * W_wp[k * 5 + tid];
    logit[tid] = a;
  }
  __syncthreads();
  if (tid == 0) {
    float mx = logit[0];
    for (int n = 1; n < 5; ++n) mx = fmaxf(mx, logit[n]);
    float s = 0.0f;
    for (int n = 0; n < 5; ++n) s += expf(logit[n] - mx);
    float lse = mx + logf(s);
    for (int n = 0; n < 5; ++n) out[n] = logit[n] - lse;
  }
}

// ---------------- Host-side launch ------------------------------------------

extern "C" void kernel_launch(void* const* d_in, const int* in_sizes, int n_in,
                              void* d_out, int out_size, void* d_ws, size_t ws_size,
                              hipStream_t stream) {
  const int*   l_tokens = (const int*)d_in[0];
  const int*   r_tokens = (const int*)d_in[1];
  const int*   left_idx = (const int*)d_in[2];
  const int*   right_idx= (const int*)d_in[3];
  const float* emb      = (const float*)d_in[4];
  const float* W_ioux   = (const float*)d_in[5];
  const float* b_ioux   = (const float*)d_in[6];
  const float* W_iouh   = (const float*)d_in[7];
  const float* b_iouh   = (const float*)d_in[8];
  // d_in[9] = W_fx, d_in[10] = b_fx : unused by the reference
  const float* W_fh     = (const float*)d_in[11];
  const float* b_fh     = (const float*)d_in[12];
  const float* W_attn   = (const float*)d_in[13];
  const float* b_attn   = (const float*)d_in[14];
  const float* W_wh     = (const float*)d_in[15];
  const float* b_wh     = (const float*)d_in[16];
  const float* W_wp     = (const float*)d_in[17];
  const float* b_wp     = (const float*)d_in[18];
  float* out = (float*)d_out;
  float* ws  = (float*)d_ws;

  // workspace layout (floats)
  size_t oH0 = 0;
  size_t oC0 = oH0 + (size_t)NNODE * MEMD;
  size_t oH1 = oC0 + (size_t)NNODE * MEMD;
  size_t oC1 = oH1 + (size_t)NNODE * MEMD;
  size_t oA  = oC1 + (size_t)NNODE * MEMD;      // 4096*300 (covers 4096*152 too)
  size_t oP  = oA  + (size_t)4096 * INDIM;      // 4096*608 (covers leaf 4096*464)
  size_t oBl = oP  + (size_t)4096 * 608;
  size_t oBb = oBl + (size_t)300 * 464;
  size_t oS  = oBb + (size_t)152 * 608;
  size_t oV  = oS  + (size_t)2 * NNODE;

  float* H0 = ws + oH0; float* C0 = ws + oC0;
  float* H1 = ws + oH1; float* C1 = ws + oC1;
  float* A  = ws + oA;  float* P  = ws + oP;
  float* Bl = ws + oBl; float* Bb = ws + oBb;
  float* S  = ws + oS;  float* V  = ws + oV;

  const int TB = 256;

  // pack weight B-matrices (zero-padded)
  pack_b_leaf<<<(300 * 464 + TB - 1) / TB, TB, 0, stream>>>(W_ioux, Bl);
  pack_b_big<<<(152 * 608 + TB - 1) / TB, TB, 0, stream>>>(W_iouh, W_fh, Bb);

  // leaves: gather + GEMM (4096x300 @ 300x450) + gates
  leaf_gather<<<(4096 * INDIM + TB - 1) / TB, TB, 0, stream>>>(l_tokens, r_tokens, emb, A);
  {
    dim3 g(4096 / 16, 464 / 16);
    wmma_gemm_f32<<<g, 32, 0, stream>>>(A, INDIM, Bl, 464, P, 464, INDIM, 4096, IOU);
  }
  leaf_combine<<<(2 * NLEAF * MEMD + TB - 1) / TB, TB, 0, stream>>>(P, b_ioux, b_iouh,
                                                                   C0, H0, C1, H1);

  // level-parallel tree scan (balanced tree over 2048 leaves -> 11 levels)
  const int lvl[11] = {1024, 512, 256, 128, 64, 32, 16, 8, 4, 2, 1};
  int off = 0;
  for (int l = 0; l < 11; ++l) {
    int M = lvl[l];
    int rowsValid = 4 * M;
    int rowsPad = (rowsValid + 15) & ~15;
    level_gather<<<(rowsPad * 152 + TB - 1) / TB, TB, 0, stream>>>(H0, H1, left_idx, right_idx,
                                                                  A, off, M, rowsPad);
    dim3 g(rowsPad / 16, 608 / 16);
    wmma_gemm_f32<<<g, 32, 0, stream>>>(A, 152, Bb, 608, P, 608, 152, rowsValid, 600);
    level_combine<<<(2 * M * MEMD + TB - 1) / TB, TB, 0, stream>>>(P, left_idx, right_idx,
                                                                  b_iouh, b_fh,
                                                                  C0, H0, C1, H1, off, M);
    off += M;
  }

  // attention collapsed to last-row / last-col
  attn_scores<<<(2 * NNODE + TB - 1) / TB, TB, 0, stream>>>(H0, H1, S);
  attn_reduce<<<2, 256, 0, stream>>>(S, H0, H1, V);
  final_mlp<<<1, 256, 0, stream>>>(H0, H1, V, W_attn, b_attn, W_wh, b_wh, W_wp, b_wp, out);
}